// SSEGDNH_52518860095898
// MI455X (gfx1250) — compile-verified
//
#include <hip/hip_runtime.h>
#include <stdint.h>

// Problem constants (fixed by setup_inputs in the reference harness)
#define LL   16384
#define HH   6
#define DD   256
#define NN   4
#define SS   4
#define KK   2
#define HD   (HH*DD)            // 1536 floats per (token) row of q/k/v
#define NF4  (HD/4)             // 384 float4 slots per row
#define ROWS (LL*KK)            // 32768 output rows
#define NB1  64                 // route blocks (256 tokens each -> 512 entries)
#define EPB  512                // entries per rank block
#define NB3  (ROWS/EPB)         // 64

typedef __attribute__((ext_vector_type(4))) float v4f;

// ---- output layout (float element offsets into d_out) ----
#define OFF_Q    ((size_t)0)
#define OFF_K    ((size_t)ROWS*HD)
#define OFF_V    ((size_t)2*(size_t)ROWS*HD)
#define OFF_GK   ((size_t)3*(size_t)ROWS*HD)
#define OFF_BETA (OFF_GK  + (size_t)ROWS*HH)
#define OFF_ES   (OFF_BETA + (size_t)ROWS*HH)
#define OFF_MW   (OFF_ES  + (size_t)LL*NN)
#define OFF_OFFS (OFF_MW  + (size_t)LL*NN)
#define OFF_SS   (OFF_OFFS + (size_t)(SS*NN+1))
#define OFF_GS   (OFF_SS  + (size_t)(SS*NN))

// ---- workspace layout (int element offsets into d_ws) ----
#define WS_BUCKET 0                      // int[ROWS]   bucket per entry (entry = tok*K + slot)
#define WS_VALUE  (WS_BUCKET + ROWS)     // float[ROWS] routing weight per entry
#define WS_BH     (WS_VALUE  + ROWS)     // int[NB1*16] per-block histograms
#define WS_BB     (WS_BH + NB1*16)       // int[NB1*16] per-block bucket bases
#define WS_SRC    (WS_BB + NB1*16)       // int[ROWS]   source token per output row
#define WS_SCALE  (WS_SRC + ROWS)        // float[ROWS] scale per output row

__device__ inline unsigned ballot32(bool p) {
#if __has_builtin(__builtin_amdgcn_ballot_w32)
  return __builtin_amdgcn_ballot_w32(p);
#else
  return (unsigned)__ballot(p);
#endif
}

// ---------------- Kernel 1: softmax + top-2 routing + block histograms -------
__global__ void k1_route(const float* __restrict__ e,
                         const long long* __restrict__ cu,
                         int* __restrict__ ws, float* __restrict__ out)
{
  __shared__ int hist[16];
  const int tid = threadIdx.x;
  if (tid < 16) hist[tid] = 0;
  __syncthreads();

  const int t = blockIdx.x * 256 + tid;
  float p[NN];
  float m = -3.402823e38f;
#pragma unroll
  for (int i = 0; i < NN; ++i) { p[i] = e[(size_t)t*NN + i]; m = fmaxf(m, p[i]); }
  float s = 0.f;
#pragma unroll
  for (int i = 0; i < NN; ++i) { p[i] = expf(p[i] - m); s += p[i]; }
  const float inv = 1.0f / s;
#pragma unroll
  for (int i = 0; i < NN; ++i) p[i] *= inv;

  // top-2, ties -> lowest index (lax.top_k semantics)
  int i0 = 0;
#pragma unroll
  for (int i = 1; i < NN; ++i) if (p[i] > p[i0]) i0 = i;
  int i1 = (i0 == 0) ? 1 : 0;
#pragma unroll
  for (int i = 0; i < NN; ++i) if (i != i0 && p[i] > p[i1]) i1 = i;

  // sample index: searchsorted(cu, t, 'right') - 1
  int smp = 0;
#pragma unroll
  for (int i = 1; i < SS; ++i) if ((long long)t >= cu[i]) smp = i;

#pragma unroll
  for (int n = 0; n < NN; ++n) {
    out[OFF_ES + (size_t)t*NN + n] = p[n];
    out[OFF_MW + (size_t)t*NN + n] = (n == i0 || n == i1) ? 1.0f : 0.0f;
  }

  const int b0 = smp*NN + i0, b1 = smp*NN + i1;
  ws[WS_BUCKET + 2*t]     = b0;
  ws[WS_BUCKET + 2*t + 1] = b1;
  ((float*)ws)[WS_VALUE + 2*t]     = p[i0];
  ((float*)ws)[WS_VALUE + 2*t + 1] = p[i1];
  atomicAdd(&hist[b0], 1);
  atomicAdd(&hist[b1], 1);
  __syncthreads();
  if (tid < 16) ws[WS_BH + blockIdx.x*16 + tid] = hist[tid];
}

// ---------------- Kernel 2: bucket totals + exclusive scans ------------------
__global__ void k2_scan(int* __restrict__ ws, float* __restrict__ out)
{
  __shared__ int tot[16], base[16];
  const int tid = threadIdx.x;
  if (tid < 16) {
    int s = 0;
    for (int blk = 0; blk < NB1; ++blk) s += ws[WS_BH + blk*16 + tid];
    tot[tid] = s;
  }
  __syncthreads();
  if (tid == 0) {
    int run = 0;
    out[OFF_OFFS] = 0.0f;
    for (int b = 0; b < 16; ++b) {
      base[b] = run;
      run += tot[b];
      out[OFF_OFFS + 1 + b] = (float)run;
      out[OFF_SS + b] = (float)tot[b];
    }
  }
  __syncthreads();
  if (tid < 16) {
    int run = base[tid];
    for (int blk = 0; blk < NB1; ++blk) {
      ws[WS_BB + blk*16 + tid] = run;
      run += ws[WS_BH + blk*16 + tid];
    }
  }
}

// ---------------- Kernel 3: stable within-bucket ranking (1 wave / block) ----
__global__ void k3_rank(int* __restrict__ ws, float* __restrict__ out)
{
  __shared__ int cnt[16];
  const int lane = threadIdx.x;       // blockDim.x == 32 (one wave32)
  const int blk  = blockIdx.x;
  if (lane < 16) cnt[lane] = ws[WS_BB + blk*16 + lane];
  __syncthreads();
  const float* wsf = (const float*)ws;

  for (int it = 0; it < EPB/32; ++it) {
    const int ent = blk*EPB + it*32 + lane;
    const int b = ws[WS_BUCKET + ent];
    unsigned my_mask = 0; int my_upd = 0;
#pragma unroll
    for (int bb = 0; bb < 16; ++bb) {
      const unsigned mm = ballot32(b == bb);
      if (b == bb)    my_mask = mm;
      if (lane == bb) my_upd  = __popc(mm);
    }
    const int rank = __popc(my_mask & ((1u << lane) - 1u));
    const int dst  = cnt[b] + rank;
    const int tok  = ent >> 1;        // KK == 2
    ws[WS_SRC + dst] = tok;
    ((float*)ws)[WS_SCALE + dst] = wsf[WS_VALUE + ent];
    out[OFF_GS + dst] = (float)tok;
    __syncthreads();
    if (lane < 16) cnt[lane] += my_upd;
    __syncthreads();
  }
}

// ---------------- Kernel 4: the big permuted gather (~0.9 GB traffic) --------
// blockDim.x == NF4 (384): each thread owns exactly one float4 slot of the row.
__global__ void k4_gather(const float* __restrict__ q, const float* __restrict__ k,
                          const float* __restrict__ v, const float* __restrict__ gk,
                          const float* __restrict__ beta,
                          const int* __restrict__ ws, float* __restrict__ out)
{
  const int r   = blockIdx.x;
  const int tid = threadIdx.x;                  // 0..383
  const int src = ws[WS_SRC + r];
  const float w = ((const float*)ws)[WS_SCALE + r];

  // v: pure copy -> CDNA5 async DMA path through LDS (ASYNCcnt-tracked);
  // issue the load early so it overlaps the q/k VALU work below.
  __shared__ v4f buf[NF4];
  const uint32_t lds = (uint32_t)(uintptr_t)(&buf[tid]); // low 32b of generic LDS ptr == LDS byte addr
  const uint64_t ga = (uint64_t)(uintptr_t)((const char*)(v + (size_t)src*HD) + (size_t)tid*16);
  asm volatile("global_load_async_to_lds_b128 %0, %1, off"
               :: "v"(lds), "v"(ga) : "memory");

  // q, k: b128 load (temporal: each source row is read exactly K=2 times ->
  // let L2 keep it), scale by routing weight, non-temporal b128 store
  // (output is write-once, never re-read).
  const v4f* qs = (const v4f*)(q + (size_t)src*HD);
  const v4f* kss = (const v4f*)(k + (size_t)src*HD);
  v4f* qo = (v4f*)(out + OFF_Q + (size_t)r*HD);
  v4f* ko = (v4f*)(out + OFF_K + (size_t)r*HD);
  v4f a = qs[tid] * w;
  __builtin_nontemporal_store(a, &qo[tid]);
  v4f b = kss[tid] * w;
  __builtin_nontemporal_store(b, &ko[tid]);

  if (tid < HH) {
    out[OFF_GK   + (size_t)r*HH + tid] = gk  [(size_t)src*HH + tid];
    out[OFF_BETA + (size_t)r*HH + tid] = beta[(size_t)src*HH + tid];
  }

  // drain the async load, then push the row back out LDS->global.
  // No trailing wait needed: s_endpgm performs an implicit wait-idle.
  asm volatile("s_wait_asynccnt 0x0" ::: "memory");
  const uint64_t gd = (uint64_t)(uintptr_t)((char*)(out + OFF_V + (size_t)r*HD) + (size_t)tid*16);
  asm volatile("global_store_async_from_lds_b128 %0, %1, off"
               :: "v"(gd), "v"(lds) : "memory");
}

extern "C" void kernel_launch(void* const* d_in, const int* in_sizes, int n_in,
                              void* d_out, int out_size, void* d_ws, size_t ws_size,
                              hipStream_t stream)
{
  const float*     q    = (const float*)d_in[0];
  const float*     k    = (const float*)d_in[1];
  const float*     v    = (const float*)d_in[2];
  const float*     gk   = (const float*)d_in[3];
  const float*     beta = (const float*)d_in[4];
  const float*     e    = (const float*)d_in[5];
  const long long* cu   = (const long long*)d_in[6];
  (void)in_sizes; (void)n_in; (void)out_size; (void)ws_size;

  int*   ws  = (int*)d_ws;
  float* out = (float*)d_out;

  k1_route <<<NB1,   256, 0, stream>>>(e, cu, ws, out);
  k2_scan  <<<1,     256, 0, stream>>>(ws, out);
  k3_rank  <<<NB3,    32, 0, stream>>>(ws, out);
  k4_gather<<<ROWS,  NF4, 0, stream>>>(q, k, v, gk, beta, ws, out);
}